// TripletLoss_71880572665884
// MI455X (gfx1250) — compile-verified
//
#include <hip/hip_runtime.h>
#include <math.h>

typedef float v2f __attribute__((ext_vector_type(2)));
typedef float v8f __attribute__((ext_vector_type(8)));

#define N 8192
#define D 128
#define MARGIN 0.3f

// ---------------------------------------------------------------------------
// Init per-row min (squared distance) to +inf bit pattern. d2 >= 0 so the
// uint bit pattern of a float is monotone -> atomicMin on u32 is a float min.
// ---------------------------------------------------------------------------
__global__ void tl_init_rowmin(unsigned* __restrict__ rowmin_bits) {
    int i = blockIdx.x * blockDim.x + threadIdx.x;
    if (i < N) rowmin_bits[i] = 0x7F800000u;  // +inf
}

// ---------------------------------------------------------------------------
// Row squared norms: one wave32 per row, float4 loads (32 lanes * 4 = 128).
// ---------------------------------------------------------------------------
__global__ __launch_bounds__(256) void tl_rowsq(const float* __restrict__ z,
                                                float* __restrict__ sq) {
    int row  = (blockIdx.x * blockDim.x + threadIdx.x) >> 5;
    int lane = threadIdx.x & 31;
    if (row >= N) return;
    const float4* p = (const float4*)(z + (size_t)row * D);
    float4 v = p[lane];
    float s = v.x * v.x + v.y * v.y + v.z * v.z + v.w * v.w;
    #pragma unroll
    for (int m = 16; m >= 1; m >>= 1) s += __shfl_xor(s, m, 32);
    if (lane == 0) sq[row] = s;
}

// ---------------------------------------------------------------------------
// Fused GEMM (z1 @ z2^T via V_WMMA_F32_16X16X4_F32) + distance epilogue +
// diagonal extraction + row-min (hard negative mining) in squared space.
//
// Grid (64,32), 256 threads = 8 waves. Block tile 256x128. Wave w owns a
// 32x128 strip: 2 A fragments x 8 B fragments = 16 f32 accumulators.
// Per K-step: 10 b64 loads feed 16 WMMAs (A reused 8x, B reused 2x).
//
// Fragment layout (ISA 7.12.2, 32-bit 16x4 A / 4x16 B):
//   lane L holds float2 at source row (L&15), K-offset k0 + 2*(L>>4)
// C/D: lane L, VGPR v -> (M = v + 8*(L>>4), N = L&15).
// ---------------------------------------------------------------------------
__global__ __launch_bounds__(256) void tl_gemm_min(
    const float* __restrict__ z1, const float* __restrict__ z2,
    const float* __restrict__ sq1, const float* __restrict__ sq2,
    unsigned* __restrict__ rowmin_bits, float* __restrict__ pos_d2) {
    const int lane   = threadIdx.x & 31;
    const int wave   = threadIdx.x >> 5;
    const int lane16 = lane & 15;
    const int khalf  = lane >> 4;          // 0 or 1
    const int rowBase = blockIdx.y * 256 + wave * 32;   // 32 rows per wave
    const int colBase = blockIdx.x * 128;

    v8f acc[2][8];
    #pragma unroll
    for (int h = 0; h < 2; ++h)
        #pragma unroll
        for (int t = 0; t < 8; ++t) acc[h][t] = (v8f){};

    const float* aPtr0 = z1 + (size_t)(rowBase +      lane16) * D + 2 * khalf;
    const float* aPtr1 = z1 + (size_t)(rowBase + 16 + lane16) * D + 2 * khalf;
    const float* bPtr  = z2 + (size_t)(colBase +      lane16) * D + 2 * khalf;

    // K loop: 128 / 4 = 32 steps; 16 WMMAs per step.
    for (int k0 = 0; k0 < D; k0 += 4) {
        v2f a0 = *(const v2f*)(aPtr0 + k0);
        v2f a1 = *(const v2f*)(aPtr1 + k0);
        #pragma unroll
        for (int t = 0; t < 8; ++t) {
            v2f b = *(const v2f*)(bPtr + (size_t)t * 16 * D + k0);
            acc[0][t] = __builtin_amdgcn_wmma_f32_16x16x4_f32(
                false, a0, false, b, (short)0, acc[0][t], false, false);
            acc[1][t] = __builtin_amdgcn_wmma_f32_16x16x4_f32(
                false, a1, false, b, (short)0, acc[1][t], false, false);
        }
    }

    // Epilogue: d2 = sq1[i] + sq2[j] - 2*G; clamp; mask diagonal; row-min
    // (in squared space; sqrt is monotone and deferred to finalize).
    #pragma unroll
    for (int h = 0; h < 2; ++h) {
        const int rb = rowBase + 16 * h;

        float s1[8];
        #pragma unroll
        for (int v = 0; v < 8; ++v) s1[v] = sq1[rb + v + 8 * khalf];

        float lanemin[8];
        #pragma unroll
        for (int v = 0; v < 8; ++v) lanemin[v] = __builtin_inff();

        #pragma unroll
        for (int t = 0; t < 8; ++t) {
            const int j   = colBase + t * 16 + lane16;
            const float s2 = sq2[j];
            #pragma unroll
            for (int v = 0; v < 8; ++v) {
                const int i = rb + v + 8 * khalf;
                float d2 = fmaxf(s1[v] + s2 - 2.0f * acc[h][t][v], 0.0f);
                if (i == j) {                 // diagonal: positive pair
                    pos_d2[i] = d2;
                    d2 = __builtin_inff();    // exclude from hard-negative min
                }
                lanemin[v] = fminf(lanemin[v], d2);
            }
        }

        // Reduce across the 16 lanes of each half (xor masks <= 8 stay
        // inside each half), then one atomic per row.
        #pragma unroll
        for (int v = 0; v < 8; ++v) {
            float m = lanemin[v];
            m = fminf(m, __shfl_xor(m, 8, 32));
            m = fminf(m, __shfl_xor(m, 4, 32));
            m = fminf(m, __shfl_xor(m, 2, 32));
            m = fminf(m, __shfl_xor(m, 1, 32));
            if (lane16 == 0) {
                const int i = rb + v + 8 * khalf;
                atomicMin(rowmin_bits + i, __float_as_uint(m));
            }
        }
    }
}

// ---------------------------------------------------------------------------
// Final: loss_i = relu(sqrt(pos2) - sqrt(min2) + margin); mean over N.
// ---------------------------------------------------------------------------
__global__ __launch_bounds__(1024) void tl_finalize(
    const float* __restrict__ pos_d2, const unsigned* __restrict__ rowmin_bits,
    float* __restrict__ out) {
    __shared__ float red[1024];
    const int tid = threadIdx.x;
    float s = 0.0f;
    for (int i = tid; i < N; i += 1024) {
        float pos = sqrtf(pos_d2[i]);
        float neg = sqrtf(__uint_as_float(rowmin_bits[i]));
        s += fmaxf(pos - neg + MARGIN, 0.0f);
    }
    red[tid] = s;
    __syncthreads();
    #pragma unroll
    for (int off = 512; off > 0; off >>= 1) {
        if (tid < off) red[tid] += red[tid + off];
        __syncthreads();
    }
    if (tid == 0) out[0] = red[0] / (float)N;
}

// ---------------------------------------------------------------------------
// Launch. Workspace layout (floats): sq1[N] | sq2[N] | rowmin_bits[N] | pos_d2[N]
// = 128 KB total.
// ---------------------------------------------------------------------------
extern "C" void kernel_launch(void* const* d_in, const int* in_sizes, int n_in,
                              void* d_out, int out_size, void* d_ws, size_t ws_size,
                              hipStream_t stream) {
    (void)in_sizes; (void)n_in; (void)out_size; (void)ws_size;
    const float* z1 = (const float*)d_in[0];
    const float* z2 = (const float*)d_in[1];
    float* ws = (float*)d_ws;
    float*    sq1    = ws;
    float*    sq2    = ws + N;
    unsigned* rowmin = (unsigned*)(ws + 2 * N);
    float*    pos    = ws + 3 * N;

    tl_init_rowmin<<<N / 256, 256, 0, stream>>>(rowmin);
    tl_rowsq<<<(N * 32) / 256, 256, 0, stream>>>(z1, sq1);
    tl_rowsq<<<(N * 32) / 256, 256, 0, stream>>>(z2, sq2);

    dim3 grid(N / 128, N / 256);
    tl_gemm_min<<<grid, 256, 0, stream>>>(z1, z2, sq1, sq2, rowmin, pos);

    tl_finalize<<<1, 1024, 0, stream>>>(pos, rowmin, (float*)d_out);
}